// TrueQuantizationLayer_48112223649805
// MI455X (gfx1250) — compile-verified
//
#include <hip/hip_runtime.h>
#include <cstdint>
#include <cstddef>

// SAR-ADC bit-serial quantizer, MI455X (gfx1250), wave32.
// out = [ q (L*P) | Q (L*P*8, [elem][bit]) | W (36) ]  (all f32)

#define NBITS   8
#define LENGTH  8192
#define PACKET  1024
#define LP      ((size_t)LENGTH * PACKET)        // 8388608 elements
#define VR_F    (1.0f / 256.0f)                  // Vdd / 2^8 (== VREF)
#define DELTA_F 1e-30f

typedef float __attribute__((ext_vector_type(4))) f4;

__global__ __launch_bounds__(256) void sar_adc_kernel(
    const float* __restrict__ x,
    const float* __restrict__ W,
    float* __restrict__ out)
{
    __shared__ f4 xs[256];  // 4 KB staging tile for x

    const int      tid = threadIdx.x;
    const size_t   e0  = ((size_t)blockIdx.x * 256u + (size_t)tid) * 4u;  // first of 4 elements
    const uint32_t gByteOff = (uint32_t)(e0 * 4u);                        // x is 33.5MB -> fits u32

    // ---- gfx1250 async copy: global x -> LDS (per-lane 16B), tracked by ASYNCcnt ----
    {
        uint32_t ldsAddr = (uint32_t)(uintptr_t)(&xs[tid]);  // low 32 bits = LDS offset
        asm volatile("global_load_async_to_lds_b128 %0, %1, %2"
                     :
                     : "v"(ldsAddr), "v"(gByteOff), "s"(x)
                     : "memory");
    }

    // ---- overlap: pull the 36 uniform weights into SGPRs and pre-scale by VR ----
    // Since VREF == VR, the scaled table serves both the leading term (W[m]*VREF)
    // and the inner terms (W[m]*VR). s_mul_f32 rounding == reference's one rounding.
    float sc[36];
#pragma unroll
    for (int i = 0; i < 36; ++i) {
        float wv = W[i];
        wv = __uint_as_float(__builtin_amdgcn_readfirstlane(__float_as_uint(wv)));
        sc[i] = wv * VR_F;
    }

    // ---- wait for our wave's async transfer, then read the staged x ----
    asm volatile("s_wait_asynccnt 0" ::: "memory");
    const f4 xv = xs[tid];   // ds_load_b128, own slot: no cross-thread hazard

    f4 qv;
    float* const Qbase = out + LP + e0 * NBITS;

#pragma unroll
    for (int e = 0; e < 4; ++e) {
        const float xe = xv[e];
        float b[NBITS];   // (Q+1)/2 in {0,1}  (exactly; ties impossible, see analysis)
        float Qf[NBITS];  // sign values in {-1,+1}
        float acc = 0.0f;
        int m = 35;       // folds to constants under full unroll

#pragma unroll
        for (int j = NBITS - 1; j >= 0; --j) {
            float bs = sc[m--];                    // W[m]*VREF (scalar constant)
#pragma unroll
            for (int k = j + 1; k < NBITS; ++k) {
                // exact: b[k] in {0,1} => fma adds exactly 0 or W[m]*VR, matching
                // reference's ((Q+1)*0.5*W[m])*VR term and its rounding.
                bs = fmaf(b[k], sc[m--], bs);
            }
            float t = (xe - bs) + DELTA_F;         // reference op order
            b[j] = (t > 0.0f) ? 1.0f : 0.0f;       // v_cmp + v_cndmask
            // exact for b in {0,1}: gives exactly -1.0f / +1.0f; packs as v_pk_fma_f32
            Qf[j] = fmaf(b[j], 2.0f, -1.0f);
            // q term is b * 2^(j-8): exact power of two, order-independent exact sum
            acc = fmaf(b[j], (float)(1 << j) * VR_F, acc);
        }
        qv[e] = acc;

        // Q for this element: 8 contiguous f32 -> two b128 stores.
        // 4 elements/thread => thread owns exactly one 128B line of Q.
        f4 lo = { Qf[0], Qf[1], Qf[2], Qf[3] };
        f4 hi = { Qf[4], Qf[5], Qf[6], Qf[7] };
        *(f4*)(Qbase + e * NBITS)     = lo;
        *(f4*)(Qbase + e * NBITS + 4) = hi;
    }

    // q: 16B per thread, contiguous across the wave (b128, fully coalesced)
    *(f4*)(out + e0) = qv;
}

__global__ void w_passthrough_kernel(const float* __restrict__ W,
                                     float* __restrict__ out)
{
    const int i = threadIdx.x;
    if (i < 36) out[LP * 9 + i] = W[i];
}

extern "C" void kernel_launch(void* const* d_in, const int* in_sizes, int n_in,
                              void* d_out, int out_size, void* d_ws, size_t ws_size,
                              hipStream_t stream)
{
    (void)in_sizes; (void)n_in; (void)out_size; (void)d_ws; (void)ws_size;

    const float* x = (const float*)d_in[0];   // [8192,1024] f32
    const float* W = (const float*)d_in[1];   // [36] f32
    float*       o = (float*)d_out;           // q | Q | W

    const int threads = 256;
    const int blocks  = (int)(LP / (threads * 4));   // 8192, exact cover

    sar_adc_kernel<<<blocks, threads, 0, stream>>>(x, W, o);
    w_passthrough_kernel<<<1, 64, 0, stream>>>(W, o);
}